// HL_LH_GAT_78151224918235
// MI455X (gfx1250) — compile-verified
//
#include <hip/hip_runtime.h>
#include <hip/hip_bf16.h>
#include <math.h>

// ---------------------------------------------------------------------------
// EGAT pipeline for MI455X (gfx1250, wave32). f32 WMMA 16x16x4 for all GEMMs.
// Sizes: N=20000 E0=320000 M=160000 E1=480000 D=64 H=2 L=6.
// GEMM register-blocks 4x4 tiles per wave (64 rows x 64 cols) so B fragments
// are amortized: 12 VMEM loads per 16 WMMAs.
// ---------------------------------------------------------------------------

typedef __attribute__((ext_vector_type(2))) float v2f;
typedef __attribute__((ext_vector_type(8))) float v8f;

#define D_N  20000
#define D_E0 320000
#define D_M  160000
#define D_E1 480000

__device__ __forceinline__ float lrelu(float x) { return x > 0.f ? x : 0.01f * x; }

// ---------------------------------------------------------------------------
// GEMM: Y[n,Cout] = X[n,K] @ W[K,Cout] (+bias). One wave per block; each wave
// computes a 64-row x 64-col macro-tile as 4x4 WMMA tiles (128 acc VGPRs).
// grid = (ceil(n/64), Cout/64). Tail rows: loads clamped (EXEC stays all-1s
// for WMMA), stores predicated.
// ---------------------------------------------------------------------------
template <int K>
__global__ __launch_bounds__(32) void gemm_wmma_k(const float* __restrict__ X,
                                                  const float* __restrict__ W,
                                                  const float* __restrict__ bias,
                                                  float* __restrict__ Y,
                                                  int n, int Cout) {
  const int lane = threadIdx.x;
  const int hi = lane >> 4;      // 0: K pair {0,1}, 1: K pair {2,3} per k-step
  const int m  = lane & 15;
  const int r0 = blockIdx.x * 64;
  const int c0 = blockIdx.y * 64;

  int rows[4];
#pragma unroll
  for (int i = 0; i < 4; ++i) {
    const int r = r0 + i * 16 + m;
    rows[i] = r < n ? r : n - 1;  // clamp for tail block (loads only)
  }

  v8f acc[4][4] = {};
#pragma unroll
  for (int kk = 0; kk < K / 4; ++kk) {
    const int k0 = kk * 4 + hi * 2;
    v2f a[4];
#pragma unroll
    for (int i = 0; i < 4; ++i)
      a[i] = *(const v2f*)(X + (size_t)rows[i] * K + k0);
    v2f b[4];
#pragma unroll
    for (int t = 0; t < 4; ++t) {
      const int col = c0 + t * 16 + m;
      b[t].x = W[(size_t)k0 * Cout + col];
      b[t].y = W[(size_t)(k0 + 1) * Cout + col];
    }
#pragma unroll
    for (int i = 0; i < 4; ++i)
#pragma unroll
      for (int t = 0; t < 4; ++t)
        acc[i][t] = __builtin_amdgcn_wmma_f32_16x16x4_f32(
            false, a[i], false, b[t], (short)0, acc[i][t], false, false);
  }

#pragma unroll
  for (int i = 0; i < 4; ++i) {
    const int rbase = r0 + i * 16 + hi * 8;
#pragma unroll
    for (int t = 0; t < 4; ++t) {
      const int col = c0 + t * 16 + m;
      const float bv = bias ? bias[col] : 0.0f;
#pragma unroll
      for (int r = 0; r < 8; ++r) {
        const int rr = rbase + r;
        if (rr < n) Y[(size_t)rr * Cout + col] = acc[i][t][r] + bv;
      }
    }
  }
}

// ---------------------------------------------------------------------------
// Fused edge kernel: per wave, 16 edges.
// Phase 1: eW = efeats[16x64] @ Wf[64x128] via WMMA -> LDS tile.
// Phase 2: lane (e,h): f = f_ni[src]+f_nj[dst]+eW+be, leaky_relu;
//          logit[e,h] = dot(f, attn[h]); ef_out[e] = f[h=0]+f[h=1] (shfl).
// Block = 128 threads (4 waves) -> 64 edges/block. grid = E/64 (exact).
// ---------------------------------------------------------------------------
__global__ __launch_bounds__(128) void edge_fused(
    const float* __restrict__ efeats, const float* __restrict__ Wf,
    const float* __restrict__ FNI, const float* __restrict__ FNJ,
    const float* __restrict__ be_l, const float* __restrict__ attn_l,
    const int* __restrict__ src, const int* __restrict__ dst,
    float* __restrict__ logits, float* __restrict__ ef_out, int E) {
  __shared__ float tile[4][16 * 132];  // 132-stride padding vs bank conflicts

  const int wave = threadIdx.x >> 5;
  const int lane = threadIdx.x & 31;
  const int ebase = (blockIdx.x * 4 + wave) * 16;
  const int hi = lane >> 4;
  const int m  = lane & 15;

  // ---- phase 1: WMMA GEMM for this wave's 16 edges ----
  {
    const float* xrow = efeats + (size_t)(ebase + m) * 64;
    v2f afrag[16];
#pragma unroll
    for (int kk = 0; kk < 16; ++kk) afrag[kk] = *(const v2f*)(xrow + kk * 4 + hi * 2);

    float* myl = &tile[wave][0];
#pragma unroll
    for (int t = 0; t < 8; ++t) {
      const int col = t * 16 + m;
      v8f acc = {};
#pragma unroll
      for (int kk = 0; kk < 16; ++kk) {
        const int k0 = kk * 4 + hi * 2;
        v2f b;
        b.x = Wf[(size_t)k0 * 128 + col];
        b.y = Wf[(size_t)(k0 + 1) * 128 + col];
        acc = __builtin_amdgcn_wmma_f32_16x16x4_f32(false, afrag[kk], false, b,
                                                    (short)0, acc, false, false);
      }
#pragma unroll
      for (int r = 0; r < 8; ++r) myl[(r + hi * 8) * 132 + col] = acc[r];
    }
  }

  // ---- phase 2: per-(edge,head) lane ----
  const int e = ebase + (lane >> 1);
  const int h = lane & 1;
  const int s = src[e], dn = dst[e];
  const float* pi = FNI + (size_t)s  * 128 + h * 64;
  const float* pj = FNJ + (size_t)dn * 128 + h * 64;
  const float* pl = &tile[wave][(lane >> 1) * 132 + h * 64];
  const float* pb = be_l  + h * 64;
  const float* pa = attn_l + h * 64;

  float logit = 0.f;
  float4 fv[16];
#pragma unroll
  for (int q = 0; q < 16; ++q) {
    const float4 a = *(const float4*)(pi + q * 4);
    const float4 b = *(const float4*)(pj + q * 4);
    const float4 w = *(const float4*)(pl + q * 4);
    const float4 c = *(const float4*)(pb + q * 4);
    const float4 at = *(const float4*)(pa + q * 4);
    float4 f;
    f.x = lrelu(a.x + b.x + w.x + c.x);
    f.y = lrelu(a.y + b.y + w.y + c.y);
    f.z = lrelu(a.z + b.z + w.z + c.z);
    f.w = lrelu(a.w + b.w + w.w + c.w);
    logit = fmaf(f.x, at.x, logit);
    logit = fmaf(f.y, at.y, logit);
    logit = fmaf(f.z, at.z, logit);
    logit = fmaf(f.w, at.w, logit);
    fv[q] = f;
  }
  logits[(size_t)e * 2 + h] = logit;

  if (ef_out) {  // head-summed edge features (f[h=0]+f[h=1])
#pragma unroll
    for (int q = 0; q < 16; ++q) {
      float4 g;
      g.x = __shfl_xor(fv[q].x, 1, 32);
      g.y = __shfl_xor(fv[q].y, 1, 32);
      g.z = __shfl_xor(fv[q].z, 1, 32);
      g.w = __shfl_xor(fv[q].w, 1, 32);
      if (h == 0) {
        float4 ssum;
        ssum.x = fv[q].x + g.x;
        ssum.y = fv[q].y + g.y;
        ssum.z = fv[q].z + g.z;
        ssum.w = fv[q].w + g.w;
        *(float4*)(ef_out + (size_t)e * 64 + q * 4) = ssum;
      }
    }
  }
}

// ---------------------------------------------------------------------------
// Segment softmax helpers + aggregation
// ---------------------------------------------------------------------------
__global__ void seg_init(float* __restrict__ emax, float* __restrict__ den, int n2) {
  int i = blockIdx.x * 256 + threadIdx.x;
  if (i < n2) { emax[i] = -INFINITY; den[i] = 0.f; }
}

__global__ void seg_max(const float* __restrict__ lg, const int* __restrict__ dst,
                        float* __restrict__ emax, int E2) {
  int i = blockIdx.x * 256 + threadIdx.x;
  if (i < E2) {
    const int e = i >> 1, h = i & 1;
    const float v = lg[i];
    float* a = &emax[(size_t)dst[e] * 2 + h];
    if (v >= 0.f) atomicMax((int*)a, __float_as_int(v));
    else          atomicMin((unsigned int*)a, __float_as_uint(v));
  }
}

__global__ void seg_expsum(const float* __restrict__ lg, const int* __restrict__ dst,
                           const float* __restrict__ emax, float* __restrict__ ex,
                           float* __restrict__ den, int E2) {
  int i = blockIdx.x * 256 + threadIdx.x;
  if (i < E2) {
    const int e = i >> 1, h = i & 1;
    const int d = dst[e] * 2 + h;
    const float v = __expf(lg[i] - emax[d]);
    ex[i] = v;
    atomicAdd(&den[d], v);
  }
}

__global__ void zero_f(float* __restrict__ p, int n) {
  int i = blockIdx.x * 256 + threadIdx.x;
  if (i < n) p[i] = 0.f;
}

// nf_out[dst] += h[src,h0]*a0 + h[src,h1]*a1 ; lane handles 4 dims of one edge.
__global__ void aggregate(const float* __restrict__ HB, const float* __restrict__ ex,
                          const float* __restrict__ den, const int* __restrict__ src,
                          const int* __restrict__ dst, float* __restrict__ out, int E) {
  int gid = blockIdx.x * 256 + threadIdx.x;
  if (gid < E * 16) {
    const int e = gid >> 4, g = gid & 15;
    const int s = src[e], d = dst[e];
    const float a0 = ex[(size_t)e * 2 + 0] / den[(size_t)d * 2 + 0];
    const float a1 = ex[(size_t)e * 2 + 1] / den[(size_t)d * 2 + 1];
    const float4 h0 = *(const float4*)(HB + (size_t)s * 128 + g * 4);
    const float4 h1 = *(const float4*)(HB + (size_t)s * 128 + 64 + g * 4);
    float* po = out + (size_t)d * 64 + g * 4;
    atomicAdd(po + 0, h0.x * a0 + h1.x * a1);
    atomicAdd(po + 1, h0.y * a0 + h1.y * a1);
    atomicAdd(po + 2, h0.z * a0 + h1.z * a1);
    atomicAdd(po + 3, h0.w * a0 + h1.w * a1);
  }
}

// ef[2m] = ef[2m+1] = np1[m]
__global__ void repeat2(const float* __restrict__ np1, float* __restrict__ ef, int E) {
  int gid = blockIdx.x * 256 + threadIdx.x;
  if (gid < E * 16) {
    const int e = gid >> 4, g = gid & 15;
    *(float4*)(ef + (size_t)e * 64 + g * 4) =
        *(const float4*)(np1 + (size_t)(e >> 1) * 64 + g * 4);
  }
}

// ---------------------------------------------------------------------------
extern "C" void kernel_launch(void* const* d_in, const int* in_sizes, int n_in,
                              void* d_out, int out_size, void* d_ws, size_t ws_size,
                              hipStream_t stream) {
  const float* node_feats = (const float*)d_in[0];
  const float* node_path  = (const float*)d_in[1];
  const float* edge_path  = (const float*)d_in[2];
  const float* Wn   = (const float*)d_in[3];
  const float* bn   = (const float*)d_in[4];
  const float* Wni  = (const float*)d_in[5];
  const float* Wnj  = (const float*)d_in[6];
  const float* Wfij = (const float*)d_in[7];
  const float* attn = (const float*)d_in[8];
  const float* be   = (const float*)d_in[9];
  const float* Wlin = (const float*)d_in[10];
  const float* blin = (const float*)d_in[11];
  const int* src0 = (const int*)d_in[12];
  const int* dst0 = (const int*)d_in[13];
  const int* src1 = (const int*)d_in[14];
  const int* dst1 = (const int*)d_in[15];

  float* ws = (float*)d_ws;
  size_t off = 0;
  float* FNI = ws + off; off += (size_t)D_M * 128;   // node src-proj (max n = M)
  float* FNJ = ws + off; off += (size_t)D_M * 128;   // node dst-proj
  float* HB  = ws + off; off += (size_t)D_M * 128;   // node value-proj (Wn + bn)
  float* LG  = ws + off; off += (size_t)D_E1 * 2;    // attention logits
  float* EX  = ws + off; off += (size_t)D_E1 * 2;    // exp(logit - max)
  float* EMX = ws + off; off += (size_t)D_M * 2;     // segment max
  float* EDN = ws + off; off += (size_t)D_M * 2;     // segment denom
  float* EFA = ws + off; off += (size_t)D_E0 * 64;   // atom edge feats
  float* NFA = ws + off; off += (size_t)D_N * 64;    // atom node feats
  float* NPB = ws + off; off += (size_t)D_M * 64;    // path node feats (np1 / npth)

  float* out_nf = (float*)d_out;
  float* out_ef = out_nf + (size_t)D_N * 64;
  float* out_np = out_ef + (size_t)D_E0 * 64;
  float* out_ep = out_np + (size_t)D_M * 64;

  auto egat = [&](int li, const float* nf, const float* ef, const int* src,
                  const int* dst, int n, int E, float* nf_out, float* ef_out) {
    const float* Wn_l  = Wn   + (size_t)li * 64 * 128;
    const float* bn_l  = bn   + (size_t)li * 128;
    const float* Wni_l = Wni  + (size_t)li * 64 * 128;
    const float* Wnj_l = Wnj  + (size_t)li * 64 * 128;
    const float* Wf_l  = Wfij + (size_t)li * 64 * 128;
    const float* at_l  = attn + (size_t)li * 128;
    const float* be_l  = be   + (size_t)li * 128;

    dim3 gg((n + 63) / 64, 2);
    gemm_wmma_k<64><<<gg, 32, 0, stream>>>(nf, Wni_l, nullptr, FNI, n, 128);
    gemm_wmma_k<64><<<gg, 32, 0, stream>>>(nf, Wnj_l, nullptr, FNJ, n, 128);
    gemm_wmma_k<64><<<gg, 32, 0, stream>>>(nf, Wn_l, bn_l, HB, n, 128);

    edge_fused<<<E / 64, 128, 0, stream>>>(ef, Wf_l, FNI, FNJ, be_l, at_l,
                                           src, dst, LG, ef_out, E);

    seg_init<<<(n * 2 + 255) / 256, 256, 0, stream>>>(EMX, EDN, n * 2);
    seg_max<<<(E * 2 + 255) / 256, 256, 0, stream>>>(LG, dst, EMX, E * 2);
    seg_expsum<<<(E * 2 + 255) / 256, 256, 0, stream>>>(LG, dst, EMX, EX, EDN, E * 2);

    zero_f<<<(n * 64 + 255) / 256, 256, 0, stream>>>(nf_out, n * 64);
    aggregate<<<E / 16, 256, 0, stream>>>(HB, EX, EDN, src, dst, nf_out, E);
  };

  // ---- block 1: gcn_1 on path graph -> np1, then repeat to atom edge feats ----
  egat(0, node_path, edge_path, src1, dst1, D_M, D_E1, NPB, nullptr);
  repeat2<<<D_E0 / 16, 256, 0, stream>>>(NPB, EFA, D_E0);

  // ---- layers 1..4 on atom graph (nf/ef updated in place; final -> d_out) ----
  egat(1, node_feats, EFA, src0, dst0, D_N, D_E0, NFA, EFA);
  egat(2, NFA, EFA, src0, dst0, D_N, D_E0, NFA, EFA);
  egat(3, NFA, EFA, src0, dst0, D_N, D_E0, NFA, EFA);
  egat(4, NFA, EFA, src0, dst0, D_N, D_E0, out_nf, out_ef);

  // ---- rebuild path node feats: x11 (= out_ef viewed [M,128]) @ Wlin + blin ----
  gemm_wmma_k<128><<<dim3((D_M + 63) / 64, 1), 32, 0, stream>>>(out_ef, Wlin, blin,
                                                                NPB, D_M, 64);

  // ---- block 2: gcn_4 on path graph -> np2, ep2 ----
  egat(5, NPB, edge_path, src1, dst1, D_M, D_E1, out_np, out_ep);
}